// EncoderBlock_88158498718112
// MI455X (gfx1250) — compile-verified
//
#include <hip/hip_runtime.h>
#include <cstddef>

// ---------------------------------------------------------------------------
// Mamba encoder block for MI455X (gfx1250), fp32 via V_WMMA_F32_16X16X4_F32.
// Each wave computes a 16xNT*16 output strip: A fragment reused across NT WMMAs.
// ---------------------------------------------------------------------------

typedef __attribute__((ext_vector_type(2))) float v2f;
typedef __attribute__((ext_vector_type(8))) float v8f;

#define BN      128
#define T_LEN   256
#define DM      128      // d_model
#define DI      256      // d_inner
#define NST     16       // d_state
#define DTR     8        // dt_rank
#define XDBL_W  48       // 40 padded to 48 (dt:0-7, B:8-23, C:24-39)
#define M_ROWS  (BN * T_LEN)   // 32768

__device__ __forceinline__ v8f wmma4(v2f a, v2f b, v8f c) {
    // D = A(16x4) * B(4x16) + C   (fp32, wave32)
    return __builtin_amdgcn_wmma_f32_16x16x4_f32(false, a, false, b,
                                                 (short)0, c, false, false);
}

__device__ __forceinline__ float silu_f(float v) {
    return v / (1.0f + __expf(-v));
}

// ---------------------------------------------------------------------------
// Generic WMMA GEMM: C[m][n] = sum_k A[m][k] * W[n][k]   (B = W^T)
// Wave computes 16 x (16*NT). Ragged N handled by clamping the weight row:
// pad columns receive finite garbage that downstream never reads.
// ---------------------------------------------------------------------------
template <int NT>
__global__ void wmma_gemm_kernel(const float* __restrict__ A, int lda,
                                 const float* __restrict__ W, int ldw, int nmax,
                                 float* __restrict__ C, int ldc,
                                 int K, int ngroups) {
    const int lane = threadIdx.x & 31;
    const int wid  = threadIdx.x >> 5;
    const int tile = blockIdx.x * 8 + wid;
    const int mt = tile / ngroups, ng = tile % ngroups;
    const int m0 = mt * 16, n0 = ng * 16 * NT;
    const int half = lane >> 4, l = lane & 15;

    const float* arow = A + (size_t)(m0 + l) * lda;
    const float* wrow[NT];
#pragma unroll
    for (int t = 0; t < NT; ++t) {
        int wn = n0 + t * 16 + l;
        if (wn >= nmax) wn = nmax - 1;   // clamp: no EXEC divergence in loop
        wrow[t] = W + (size_t)wn * ldw;
    }

    v8f zero = {};
    v8f acc[NT];
#pragma unroll
    for (int t = 0; t < NT; ++t) acc[t] = zero;

#pragma unroll 2
    for (int k0 = 0; k0 < K; k0 += 4) {
        const int ka = k0 + half * 2;
        v2f a; a.x = arow[ka]; a.y = arow[ka + 1];
#pragma unroll
        for (int t = 0; t < NT; ++t) {
            v2f b; b.x = wrow[t][ka]; b.y = wrow[t][ka + 1];
            acc[t] = wmma4(a, b, acc[t]);
        }
    }
#pragma unroll
    for (int t = 0; t < NT; ++t) {
        const int col = n0 + t * 16 + l;
#pragma unroll
        for (int v = 0; v < 8; ++v) {
            const int row = m0 + v + half * 8;
            C[(size_t)row * ldc + col] = acc[t][v];
        }
    }
}

// ---------------------------------------------------------------------------
// Depthwise causal conv (K=4, left pad 3) + bias + SiLU.
// Reads xc half of xz (B,T,512); writes xc_act (B,T,256).
// ---------------------------------------------------------------------------
__global__ void conv_silu_kernel(const float* __restrict__ xz,
                                 const float* __restrict__ cw,
                                 const float* __restrict__ cb,
                                 float* __restrict__ xc) {
    const int i = blockIdx.x * blockDim.x + threadIdx.x;
    if (i >= BN * T_LEN * DI) return;
    const int d = i % DI;
    const int t = (i / DI) % T_LEN;
    const int b = i / (DI * T_LEN);
    float acc = cb[d];
#pragma unroll
    for (int j = 0; j < 4; ++j) {
        const int tj = t - 3 + j;
        if (tj >= 0)
            acc += cw[d * 4 + j] * xz[((size_t)b * T_LEN + tj) * (2 * DI) + d];
    }
    xc[i] = silu_f(acc);
}

// ---------------------------------------------------------------------------
// Selective scan. Block = one batch (256 threads = one d-channel each).
// State h[16] in registers; shared dt/B/C staged via LDS, 16 steps per chunk.
// dt_proj + softplus fused. Writes ys (B,T,256).
// ---------------------------------------------------------------------------
#define TCHUNK 16
__global__ void scan_kernel(const float* __restrict__ xdbl,   // (B,T,48)
                            const float* __restrict__ xc,     // (B,T,256)
                            const float* __restrict__ dtw,    // (256,8)
                            const float* __restrict__ dtb,    // (256)
                            const float* __restrict__ A_log,  // (256,16)
                            float* __restrict__ ys) {         // (B,T,256)
    const int b = blockIdx.x;
    const int d = threadIdx.x;

    float Arow[NST];
#pragma unroll
    for (int n = 0; n < NST; ++n) Arow[n] = -__expf(A_log[d * NST + n]);
    float w[DTR];
#pragma unroll
    for (int r = 0; r < DTR; ++r) w[r] = dtw[d * DTR + r];
    const float bias = dtb[d];

    float h[NST];
#pragma unroll
    for (int n = 0; n < NST; ++n) h[n] = 0.0f;

    __shared__ float sh[TCHUNK * 40];
    const float* xd  = xdbl + (size_t)b * T_LEN * XDBL_W;
    const float* xcb = xc   + (size_t)b * T_LEN * DI;
    float*       ysb = ys   + (size_t)b * T_LEN * DI;

    for (int t0 = 0; t0 < T_LEN; t0 += TCHUNK) {
        for (int i = threadIdx.x; i < TCHUNK * 40; i += blockDim.x) {
            const int tt = i / 40, cc = i % 40;
            sh[i] = xd[(size_t)(t0 + tt) * XDBL_W + cc];
        }
        __syncthreads();
        for (int tt = 0; tt < TCHUNK; ++tt) {
            const float* row = &sh[tt * 40];
            float acc = bias;
#pragma unroll
            for (int r = 0; r < DTR; ++r) acc += row[r] * w[r];
            const float delta = (acc > 20.0f) ? acc : log1pf(__expf(acc));
            const float u  = xcb[(size_t)(t0 + tt) * DI + d];
            const float du = delta * u;
            float y = 0.0f;
#pragma unroll
            for (int n = 0; n < NST; ++n) {
                const float dA = __expf(delta * Arow[n]);
                h[n] = dA * h[n] + du * row[8 + n];
                y += h[n] * row[24 + n];
            }
            ysb[(size_t)(t0 + tt) * DI + d] = y;
        }
        __syncthreads();
    }
}

// ---------------------------------------------------------------------------
// y = (ys + xc * D) * silu(z); z from second half of xz. In-place into ys.
// ---------------------------------------------------------------------------
__global__ void gate_kernel(float* __restrict__ ys,
                            const float* __restrict__ xc,
                            const float* __restrict__ Dv,
                            const float* __restrict__ xz) {
    const int i = blockIdx.x * blockDim.x + threadIdx.x;
    if (i >= BN * T_LEN * DI) return;
    const int d = i % DI;
    const int t = (i / DI) % T_LEN;
    const int b = i / (DI * T_LEN);
    const float z = xz[((size_t)b * T_LEN + t) * (2 * DI) + DI + d];
    ys[i] = (ys[i] + xc[i] * Dv[d]) * silu_f(z);
}

// ---------------------------------------------------------------------------
// out_proj GEMM: h_mid = Y(32768x256) * out_proj_w^T (128x256),
// stored into time-padded HP (B, T+2, 128) at rows t+1. Wave = 16x64 strip.
// ---------------------------------------------------------------------------
__global__ void outproj_kernel(const float* __restrict__ Y,
                               const float* __restrict__ W,
                               float* __restrict__ HP) {
    const int lane = threadIdx.x & 31;
    const int wid  = threadIdx.x >> 5;
    const int tile = blockIdx.x * 8 + wid;   // 2048 m-tiles * 2 n-groups
    const int mt = tile >> 1, ng = tile & 1;
    const int m0 = mt * 16, n0 = ng * 64;
    const int half = lane >> 4, l = lane & 15;

    const float* arow = Y + (size_t)(m0 + l) * DI;
    const float* wrow[4];
#pragma unroll
    for (int t = 0; t < 4; ++t) wrow[t] = W + (size_t)(n0 + t * 16 + l) * DI;

    v8f zero = {};
    v8f acc[4];
#pragma unroll
    for (int t = 0; t < 4; ++t) acc[t] = zero;

#pragma unroll 2
    for (int k0 = 0; k0 < DI; k0 += 4) {
        const int ka = k0 + half * 2;
        v2f a; a.x = arow[ka]; a.y = arow[ka + 1];
#pragma unroll
        for (int t = 0; t < 4; ++t) {
            v2f b; b.x = wrow[t][ka]; b.y = wrow[t][ka + 1];
            acc[t] = wmma4(a, b, acc[t]);
        }
    }
#pragma unroll
    for (int t = 0; t < 4; ++t) {
        const int col = n0 + t * 16 + l;
#pragma unroll
        for (int v = 0; v < 8; ++v) {
            const int m  = m0 + v + half * 8;
            const int bb = m >> 8;
            const int tt = m & 255;
            HP[((size_t)bb * (T_LEN + 2) + tt + 1) * DM + col] = acc[t][v];
        }
    }
}

// Zero the two pad rows (tp=0, tp=T+1) of HP per batch.
__global__ void zeropad_kernel(float* __restrict__ HP) {
    const int i = blockIdx.x * blockDim.x + threadIdx.x;
    if (i >= BN * 2 * DM) return;
    const int c     = i % DM;
    const int which = (i / DM) & 1;
    const int b     = i / (2 * DM);
    const int row   = b * (T_LEN + 2) + (which ? (T_LEN + 1) : 0);
    HP[(size_t)row * DM + c] = 0.0f;
}

// ---------------------------------------------------------------------------
// Strided (stride 2) dense conv as 3 accumulated WMMA GEMMs over HP:
// out[b][to][co] = sum_k sum_ci HP[b][2*to+k][ci] * down_w[co][ci][k] + down_b.
// Wave = 16 t_out x 64 co strip.
// ---------------------------------------------------------------------------
__global__ void down_kernel(const float* __restrict__ HP,
                            const float* __restrict__ Wd,   // (128,128,3)
                            const float* __restrict__ bd,
                            float* __restrict__ OUT) {      // (B,128,128)
    const int lane = threadIdx.x & 31;
    const int wid  = threadIdx.x >> 5;
    const int tile = blockIdx.x * 8 + wid;   // 128 batches * 8 mt * 2 ng = 2048
    const int b  = tile >> 4;
    const int mt = (tile >> 1) & 7, ng = tile & 1;
    const int to0 = mt * 16, n0 = ng * 64;
    const int half = lane >> 4, l = lane & 15;

    const float* hb = HP + (size_t)b * (T_LEN + 2) * DM;
    const float* wb[4];
#pragma unroll
    for (int t = 0; t < 4; ++t) wb[t] = Wd + (size_t)(n0 + t * 16 + l) * DM * 3;

    v8f zero = {};
    v8f acc[4];
#pragma unroll
    for (int t = 0; t < 4; ++t) acc[t] = zero;

#pragma unroll
    for (int k = 0; k < 3; ++k) {
        const float* arow = hb + (size_t)(2 * (to0 + l) + k) * DM;
#pragma unroll 2
        for (int ci0 = 0; ci0 < DM; ci0 += 4) {
            const int ka = ci0 + half * 2;
            v2f a; a.x = arow[ka]; a.y = arow[ka + 1];
#pragma unroll
            for (int t = 0; t < 4; ++t) {
                v2f bv;
                bv.x = wb[t][(size_t)ka * 3 + k];
                bv.y = wb[t][(size_t)(ka + 1) * 3 + k];
                acc[t] = wmma4(a, bv, acc[t]);
            }
        }
    }
#pragma unroll
    for (int t = 0; t < 4; ++t) {
        const int co   = n0 + t * 16 + l;
        const float bias = bd[co];
#pragma unroll
        for (int v = 0; v < 8; ++v) {
            const int to = to0 + v + half * 8;
            OUT[((size_t)b * 128 + to) * DM + co] = acc[t][v] + bias;
        }
    }
}

// ---------------------------------------------------------------------------
// LayerNorm over channel dim (128). One block per row.
// ---------------------------------------------------------------------------
__global__ void ln_kernel(const float* __restrict__ in,
                          const float* __restrict__ g,
                          const float* __restrict__ bta,
                          float* __restrict__ out) {
    const int row = blockIdx.x;
    const int c   = threadIdx.x;
    const float v = in[(size_t)row * DM + c];
    __shared__ float s1[DM], s2[DM];
    s1[c] = v; s2[c] = v * v;
    __syncthreads();
    for (int off = DM / 2; off > 0; off >>= 1) {
        if (c < off) { s1[c] += s1[c + off]; s2[c] += s2[c + off]; }
        __syncthreads();
    }
    const float mu  = s1[0] * (1.0f / DM);
    const float var = s2[0] * (1.0f / DM) - mu * mu;
    const float r   = rsqrtf(var + 1e-5f);
    out[(size_t)row * DM + c] = (v - mu) * r * g[c] + bta[c];
}

// ---------------------------------------------------------------------------
extern "C" void kernel_launch(void* const* d_in, const int* in_sizes, int n_in,
                              void* d_out, int out_size, void* d_ws, size_t ws_size,
                              hipStream_t stream) {
    const float* x        = (const float*)d_in[0];
    const float* in_w     = (const float*)d_in[1];   // (512,128)
    const float* conv_w   = (const float*)d_in[2];   // (256,1,4)
    const float* conv_b   = (const float*)d_in[3];
    const float* xproj_w  = (const float*)d_in[4];   // (40,256)
    const float* dt_w     = (const float*)d_in[5];   // (256,8)
    const float* dt_b     = (const float*)d_in[6];
    const float* A_log    = (const float*)d_in[7];   // (256,16)
    const float* Dv       = (const float*)d_in[8];
    const float* out_w    = (const float*)d_in[9];   // (128,256)
    const float* down_w   = (const float*)d_in[10];  // (128,128,3)
    const float* down_b   = (const float*)d_in[11];
    const float* ln_g     = (const float*)d_in[12];
    const float* ln_b     = (const float*)d_in[13];
    float* out = (float*)d_out;

    // workspace layout (floats)
    float* ws       = (float*)d_ws;
    float* buf_xz   = ws;                                   // 128*256*512
    float* buf_xc   = buf_xz  + (size_t)M_ROWS * 2 * DI;    // 128*256*256
    float* buf_xdbl = buf_xc  + (size_t)M_ROWS * DI;        // 128*256*48
    float* buf_ys   = buf_xdbl + (size_t)M_ROWS * XDBL_W;   // 128*256*256
    // xz is dead after gate_kernel: alias pad/down buffers into it
    float* buf_hpad = buf_xz;                               // 128*258*128
    float* buf_down = buf_xz + (size_t)BN * (T_LEN + 2) * DM; // 128*128*128

    // 1) in_proj: xz = x * in_proj_w^T  (M=32768, K=128, N=512)
    //    NT=4 -> 8 n-groups of 64; 2048 m-tiles * 8 groups / 8 waves = 2048 blocks
    wmma_gemm_kernel<4><<<2048, 256, 0, stream>>>(
        x, DM, in_w, DM, 2 * DI, buf_xz, 2 * DI, DM, 8);

    // 2) depthwise causal conv + SiLU
    conv_silu_kernel<<<(M_ROWS * DI + 255) / 256, 256, 0, stream>>>(
        buf_xz, conv_w, conv_b, buf_xc);

    // 3) x_proj: x_dbl = xc * x_proj_w^T  (N=40 padded to 48, NT=3, 1 group)
    wmma_gemm_kernel<3><<<2048 / 8, 256, 0, stream>>>(
        buf_xc, DI, xproj_w, DI, 40, buf_xdbl, XDBL_W, DI, 1);

    // 4) selective scan (dt_proj + softplus fused)
    scan_kernel<<<BN, DI, 0, stream>>>(buf_xdbl, buf_xc, dt_w, dt_b, A_log, buf_ys);

    // 5) y = (ys + xc*D) * silu(z)
    gate_kernel<<<(M_ROWS * DI + 255) / 256, 256, 0, stream>>>(
        buf_ys, buf_xc, Dv, buf_xz);

    // 6) zero pad rows, out_proj into padded layout (2048 mt * 2 ng / 8 = 512)
    zeropad_kernel<<<(BN * 2 * DM + 255) / 256, 256, 0, stream>>>(buf_hpad);
    outproj_kernel<<<512, 256, 0, stream>>>(buf_ys, out_w, buf_hpad);

    // 7) strided down-conv (3 accumulated WMMA GEMMs) + bias (2048/8 = 256)
    down_kernel<<<256, 256, 0, stream>>>(buf_hpad, down_w, down_b, buf_down);

    // 8) LayerNorm -> first output
    ln_kernel<<<BN * 128, DM, 0, stream>>>(buf_down, ln_g, ln_b, out);

    // 9) x_skip -> second output
    hipMemcpyAsync(out + (size_t)BN * 128 * DM, x,
                   (size_t)BN * T_LEN * DM * sizeof(float),
                   hipMemcpyDeviceToDevice, stream);
}